// Manager_18708877541654
// MI455X (gfx1250) — compile-verified
//
#include <hip/hip_runtime.h>
#include <hip/hip_bf16.h>
#include <math.h>

// Sizes from the reference
#define Bsz   64
#define Nn    64
#define H2    1024
#define VOCAB 32000

typedef __attribute__((ext_vector_type(2))) float v2f;
typedef __attribute__((ext_vector_type(8))) float v8f;

// ---------------------------------------------------------------------------
// Kernel 1: prior = log_softmax(einsum("bh,bnh->bn", x, K))
// One block per batch b. x[b] staged in LDS. 4 lanes per row-dot.
// ---------------------------------------------------------------------------
__global__ __launch_bounds__(256) void prior_kernel(
    const float* __restrict__ x, const float* __restrict__ K,
    float* __restrict__ prior)
{
    __shared__ float xs[H2];
    __shared__ float sc[Nn];
    __shared__ float red;
    const int b   = blockIdx.x;
    const int tid = threadIdx.x;

    for (int i = tid; i < H2; i += 256) xs[i] = x[b * H2 + i];
    __syncthreads();

    const int n = tid >> 2;       // row 0..63
    const int j = tid & 3;        // sub-lane 0..3
    const float* Kr = K + ((size_t)b * Nn + n) * H2;
    float p = 0.f;
    for (int h = j; h < H2; h += 4) p += Kr[h] * xs[h];
    p += __shfl_xor(p, 1);
    p += __shfl_xor(p, 2);
    if (j == 0) sc[n] = p;
    __syncthreads();

    if (tid == 0) {
        float m = sc[0];
        for (int i = 1; i < Nn; ++i) m = fmaxf(m, sc[i]);
        float s = 0.f;
        for (int i = 0; i < Nn; ++i) s += expf(sc[i] - m);
        red = m + logf(s);               // logsumexp
    }
    __syncthreads();
    if (tid < Nn) prior[b * Nn + tid] = sc[tid] - red;
}

// ---------------------------------------------------------------------------
// Kernel 2: response = cat(x,y) @ W1^T + b1     (M=64, K=2048, N=1024)
// WMMA f32 16x16x4. 8 waves/block: 4 M-tiles x 2 N-tiles -> 64x32 per block.
// K-loop split into x-half / y-half: no per-iteration operand select.
// ---------------------------------------------------------------------------
__global__ __launch_bounds__(256) void gemm_response(
    const float* __restrict__ x, const float* __restrict__ y,
    const float* __restrict__ W1, const float* __restrict__ b1,
    float* __restrict__ response)
{
    const int lane = threadIdx.x & 31;
    const int wave = threadIdx.x >> 5;
    const int m0   = (wave & 3) * 16;
    const int n0   = blockIdx.x * 32 + (wave >> 2) * 16;
    const int half = lane >> 4;          // 0: lanes 0-15, 1: lanes 16-31
    const int l15  = lane & 15;
    const int arow = m0 + l15;           // A-matrix row   (batch index)
    const int bcol = n0 + l15;           // B-matrix col   (W1 row)

    const float* aptr = x + (size_t)arow * H2 + half * 2;
    const float* bptr = W1 + (size_t)bcol * (2 * H2) + half * 2;

    v8f acc = {};
    // ---- x half: k = 0 .. H2-1 ----
    for (int k0 = 0; k0 < H2; k0 += 64) {
        __builtin_prefetch(bptr + 256, 0, 1);   // global_prefetch_b8, 1KB ahead
#pragma unroll
        for (int u = 0; u < 16; ++u) {
            v2f a  = *(const v2f*)(aptr + u * 4);
            v2f bb = *(const v2f*)(bptr + u * 4);
            acc = __builtin_amdgcn_wmma_f32_16x16x4_f32(
                      false, a, false, bb, (short)0, acc, false, false);
        }
        aptr += 64; bptr += 64;
    }
    // ---- y half: k = H2 .. 2*H2-1 ----
    aptr = y + (size_t)arow * H2 + half * 2;
    for (int k0 = 0; k0 < H2; k0 += 64) {
        __builtin_prefetch(bptr + 256, 0, 1);
#pragma unroll
        for (int u = 0; u < 16; ++u) {
            v2f a  = *(const v2f*)(aptr + u * 4);
            v2f bb = *(const v2f*)(bptr + u * 4);
            acc = __builtin_amdgcn_wmma_f32_16x16x4_f32(
                      false, a, false, bb, (short)0, acc, false, false);
        }
        aptr += 64; bptr += 64;
    }

    const int n = n0 + l15;
    const float bias = b1[n];
#pragma unroll
    for (int r = 0; r < 8; ++r) {
        const int m = m0 + r + half * 8;
        response[(size_t)m * H2 + n] = acc[r] + bias;
    }
}

// ---------------------------------------------------------------------------
// Kernel 3: posterior, gumbel-softmax k_idx, and k_i = k_idx @ K[b]
// One block per batch b.
// ---------------------------------------------------------------------------
__global__ __launch_bounds__(256) void posterior_kernel(
    const float* __restrict__ response, const float* __restrict__ K,
    const float* __restrict__ gu,
    float* __restrict__ posterior, float* __restrict__ k_i)
{
    __shared__ float rs[H2];
    __shared__ float sc[Nn];   // posterior logits
    __shared__ float w[Nn];    // gumbel logits -> k_idx weights
    __shared__ float red[4];
    const int b   = blockIdx.x;
    const int tid = threadIdx.x;

    for (int i = tid; i < H2; i += 256) rs[i] = response[b * H2 + i];
    __syncthreads();

    const int n = tid >> 2;
    const int j = tid & 3;
    const float* Kr = K + ((size_t)b * Nn + n) * H2;
    float p = 0.f;
    for (int h = j; h < H2; h += 4) p += Kr[h] * rs[h];
    p += __shfl_xor(p, 1);
    p += __shfl_xor(p, 2);
    if (j == 0) sc[n] = p;
    __syncthreads();

    if (tid < Nn) {
        const float u = gu[b * Nn + tid];
        const float g = -logf(-logf(u + 1e-10f) + 1e-10f);
        w[tid] = (sc[tid] + g) / 0.8f;   // TEMPERATURE
    }
    __syncthreads();

    if (tid == 0) {
        float m = sc[0], mt = w[0];
        for (int i = 1; i < Nn; ++i) { m = fmaxf(m, sc[i]); mt = fmaxf(mt, w[i]); }
        float s = 0.f, st = 0.f;
        for (int i = 0; i < Nn; ++i) { s += expf(sc[i] - m); st += expf(w[i] - mt); }
        red[0] = m; red[1] = s; red[2] = mt; red[3] = st;
    }
    __syncthreads();

    if (tid < Nn) {
        posterior[b * Nn + tid] = expf(sc[tid] - red[0]) / red[1];
        w[tid]                  = expf(w[tid]  - red[2]) / red[3];
    }
    __syncthreads();

    // k_i[b][h] = sum_n w[n] * K[b][n][h]  (each thread owns 4 h-columns)
#pragma unroll
    for (int i = 0; i < 4; ++i) {
        const int h = tid + 256 * i;
        float acc = 0.f;
        for (int nn = 0; nn < Nn; ++nn)
            acc += w[nn] * K[((size_t)b * Nn + nn) * H2 + h];
        k_i[(size_t)b * H2 + h] = acc;
    }
}

// ---------------------------------------------------------------------------
// Kernel 4: raw logits = k_i @ Wk^T + bk   (M=64, K=1024, N=32000)
// WMMA f32 16x16x4, chunked K loop with one unconditional prefetch per chunk.
// ---------------------------------------------------------------------------
__global__ __launch_bounds__(256) void gemm_vocab(
    const float* __restrict__ k_i, const float* __restrict__ Wk,
    const float* __restrict__ bk, float* __restrict__ logits)
{
    const int lane = threadIdx.x & 31;
    const int wave = threadIdx.x >> 5;
    const int m0   = (wave & 3) * 16;
    const int n0   = blockIdx.x * 32 + (wave >> 2) * 16;
    const int half = lane >> 4;
    const int l15  = lane & 15;
    const int arow = m0 + l15;
    const int bcol = n0 + l15;

    const float* aptr = k_i + (size_t)arow * H2 + half * 2;
    const float* bptr = Wk + (size_t)bcol * H2 + half * 2;

    v8f acc = {};
    for (int k0 = 0; k0 < H2; k0 += 64) {
        __builtin_prefetch(bptr + 256, 0, 1);   // stream Wk 1KB ahead
#pragma unroll
        for (int u = 0; u < 16; ++u) {
            v2f a  = *(const v2f*)(aptr + u * 4);
            v2f bb = *(const v2f*)(bptr + u * 4);
            acc = __builtin_amdgcn_wmma_f32_16x16x4_f32(
                      false, a, false, bb, (short)0, acc, false, false);
        }
        aptr += 64; bptr += 64;
    }

    const int nc = n0 + l15;
    const float bias = bk[nc];
#pragma unroll
    for (int r = 0; r < 8; ++r) {
        const int m = m0 + r + half * 8;
        logits[(size_t)m * VOCAB + nc] = acc[r] + bias;
    }
}

// ---------------------------------------------------------------------------
// Kernel 5: in-place log_softmax over each 32000-wide row (one block per row)
// Online (m,s) per thread, LDS tree combine, then normalize pass.
// ---------------------------------------------------------------------------
__global__ __launch_bounds__(1024) void lsm_kernel(float* __restrict__ logits)
{
    __shared__ float sm[1024];
    __shared__ float ss[1024];
    const int tid = threadIdx.x;
    float* row = logits + (size_t)blockIdx.x * VOCAB;

    float m = -3.402823466e38f, s = 0.f;
    for (int i = tid; i < VOCAB; i += 1024) {
        const float v = row[i];
        if (v > m) { s = s * expf(m - v) + 1.f; m = v; }
        else       { s += expf(v - m); }
    }
    sm[tid] = m; ss[tid] = s;
    __syncthreads();
    for (int off = 512; off > 0; off >>= 1) {
        if (tid < off) {
            const float m2 = sm[tid + off], s2 = ss[tid + off];
            const float M  = fmaxf(sm[tid], m2);
            ss[tid] = ss[tid] * expf(sm[tid] - M) + s2 * expf(m2 - M);
            sm[tid] = M;
        }
        __syncthreads();
    }
    const float lse = sm[0] + logf(ss[0]);
    for (int i = tid; i < VOCAB; i += 1024) row[i] -= lse;
}

// ---------------------------------------------------------------------------
extern "C" void kernel_launch(void* const* d_in, const int* in_sizes, int n_in,
                              void* d_out, int out_size, void* d_ws, size_t ws_size,
                              hipStream_t stream)
{
    const float* x  = (const float*)d_in[0];
    const float* y  = (const float*)d_in[1];
    const float* K  = (const float*)d_in[2];
    const float* gu = (const float*)d_in[3];
    const float* W1 = (const float*)d_in[4];
    const float* b1 = (const float*)d_in[5];
    const float* Wk = (const float*)d_in[6];
    const float* bk = (const float*)d_in[7];

    float* out       = (float*)d_out;
    float* prior     = out;                        // B*N      = 4096
    float* posterior = out + Bsz * Nn;             // B*N      = 4096
    float* k_i       = out + 2 * Bsz * Nn;         // B*H2     = 65536
    float* k_logits  = out + 2 * Bsz * Nn + Bsz * H2;  // B*V  = 2048000
    // Stage `response` (256KB) in the not-yet-written k_logits region:
    // kernel 4 only reads k_i/Wk and then overwrites this region with logits.
    float* response  = k_logits;

    prior_kernel    <<<Bsz, 256, 0, stream>>>(x, K, prior);
    gemm_response   <<<H2 / 32, 256, 0, stream>>>(x, y, W1, b1, response);
    posterior_kernel<<<Bsz, 256, 0, stream>>>(response, K, gu, posterior, k_i);
    gemm_vocab      <<<VOCAB / 32, 256, 0, stream>>>(k_i, Wk, bk, k_logits);
    lsm_kernel      <<<Bsz, 1024, 0, stream>>>(k_logits);
}